// GATEAU_21036749816021
// MI455X (gfx1250) — compile-verified
//
#include <hip/hip_runtime.h>

#define D 128
#define LDSPITCH 132   // 128 + 4 pad -> conflict-free WMMA A reads, 16B-aligned rows (528B)

typedef float v2f __attribute__((ext_vector_type(2)));
typedef float v8f __attribute__((ext_vector_type(8)));

// ---- order-preserving float<->uint encoding for atomicMax on floats ----
__device__ __forceinline__ unsigned enc_f32(float f) {
    unsigned u = __float_as_uint(f);
    return (u & 0x80000000u) ? ~u : (u | 0x80000000u);
}
__device__ __forceinline__ float dec_f32(unsigned e) {
    unsigned u = (e & 0x80000000u) ? (e & 0x7FFFFFFFu) : ~e;
    return __uint_as_float(u);
}
#define ENC_NEG_INF 0x007FFFFFu   // enc_f32(-inf)

// ---- CDNA5 async global->LDS staging (ASYNCcnt-tracked) ----
__device__ __forceinline__ void async_ld_b128(unsigned lds_byte_off, const float* g) {
    asm volatile("global_load_async_to_lds_b128 %0, %1, off"
                 :: "v"(lds_byte_off), "v"(g) : "memory");
}
__device__ __forceinline__ void wait_asynccnt0() {
    asm volatile("s_wait_asynccnt 0" ::: "memory");
}

// ---------------------------------------------------------------- init ----
__global__ __launch_bounds__(256) void gnn_init(float* __restrict__ nodes_new,
                                                unsigned* __restrict__ seg_max,
                                                float* __restrict__ denom,
                                                int Nn, int ND) {
    int idx = blockIdx.x * blockDim.x + threadIdx.x;
    if (idx < ND) nodes_new[idx] = 0.0f;
    if (idx < Nn) { seg_max[idx] = ENC_NEG_INF; denom[idx] = 0.0f; }
}

// ------------------------------------------- node projections (3 GEMMs) ----
// One wave per 16x16 output tile; A loads shared across 3 weight matrices.
__global__ __launch_bounds__(32) void gnn_node_proj(
    const float* __restrict__ nodes,
    const float* __restrict__ Ws1, const float* __restrict__ bs1,
    const float* __restrict__ Wr,  const float* __restrict__ br,
    const float* __restrict__ Wm,  const float* __restrict__ bm,
    float* __restrict__ Ps1, float* __restrict__ Pr, float* __restrict__ Pm,
    int Nn)
{
    const int lane = threadIdx.x;
    const int half = lane >> 4, l15 = lane & 15;
    const int row0 = blockIdx.x * 16;
    const int col0 = blockIdx.y * 16;
    if (row0 + 16 > Nn) return;
    const int kk = half * 2;

    v8f c1 = {}; v8f c2 = {}; v8f c3 = {};
    const float* arow = nodes + (size_t)(row0 + l15) * D + kk;
    const int colg = col0 + l15;

    for (int k = 0; k < D; k += 4) {
        v2f a = *(const v2f*)(arow + k);
        const int kr = k + kk;
        v2f b1, b2, b3;
        b1.x = Ws1[(size_t)kr * D + colg];  b1.y = Ws1[(size_t)(kr + 1) * D + colg];
        b2.x = Wr [(size_t)kr * D + colg];  b2.y = Wr [(size_t)(kr + 1) * D + colg];
        b3.x = Wm [(size_t)kr * D + colg];  b3.y = Wm [(size_t)(kr + 1) * D + colg];
        c1 = __builtin_amdgcn_wmma_f32_16x16x4_f32(false, a, false, b1, (short)0, c1, false, false);
        c2 = __builtin_amdgcn_wmma_f32_16x16x4_f32(false, a, false, b2, (short)0, c2, false, false);
        c3 = __builtin_amdgcn_wmma_f32_16x16x4_f32(false, a, false, b3, (short)0, c3, false, false);
    }
    const float vb1 = bs1[colg], vb2 = br[colg], vb3 = bm[colg];
#pragma unroll
    for (int v = 0; v < 8; ++v) {
        const size_t r = (size_t)(row0 + half * 8 + v);
        Ps1[r * D + colg] = c1[v] + vb1;
        Pr [r * D + colg] = c2[v] + vb2;
        Pm [r * D + colg] = c3[v] + vb3;
    }
}

// --------------------- fused edge GEMM + combine + attention logits ----
// Block: 8 waves handle a 32-edge x 128-col tile. Wave ty -> 16-col slice,
// 32 rows via two stacked accumulators that SHARE every B fetch.
__global__ __launch_bounds__(256) void gnn_edge(
    const float* __restrict__ edges,
    const float* __restrict__ We, const float* __restrict__ be,
    const float* __restrict__ Ps1, const float* __restrict__ Pr,
    const int* __restrict__ senders, const int* __restrict__ receivers,
    const float* __restrict__ Watt, const float* __restrict__ batt,
    float* __restrict__ edges_new, float* __restrict__ att_out,
    unsigned* __restrict__ seg_max, int E)
{
    __shared__ float ldsA[32 * LDSPITCH];
    __shared__ float lds_att[32];
    __shared__ int s_idx[32];
    __shared__ int r_idx[32];

    const int tx = threadIdx.x, ty = threadIdx.y;
    const int tid = ty * 32 + tx;
    const int e0 = blockIdx.x * 32;

    if (tid < 32)                 { s_idx[tid] = senders[e0 + tid]; lds_att[tid] = 0.0f; }
    else if (tid < 64)            { r_idx[tid - 32] = receivers[e0 + tid - 32]; }

    // async-stage the 32x128 edge tile into LDS (1024 x b128, ASYNCcnt path)
    const unsigned lds_base = (unsigned)(uintptr_t)(&ldsA[0]);
#pragma unroll
    for (int rep = 0; rep < 4; ++rep) {
        const int i4  = rep * 256 + tid;        // 1024 float4 total
        const int row = i4 >> 5;                // 32 float4 per row
        const int c4  = (i4 & 31) * 4;
        async_ld_b128(lds_base + (unsigned)(row * LDSPITCH + c4) * 4u,
                      &edges[(size_t)(e0 + row) * D + c4]);
    }
    wait_asynccnt0();
    __syncthreads();

    const int half = tx >> 4, l15 = tx & 15, kk = half * 2;
    const int col0 = ty * 16;
    const int colg = col0 + l15;

    v8f acc0 = {};   // rows 0..15
    v8f acc1 = {};   // rows 16..31
    for (int k = 0; k < D; k += 4) {
        v2f a0 = *(const v2f*)&ldsA[ l15       * LDSPITCH + k + kk];
        v2f a1 = *(const v2f*)&ldsA[(l15 + 16) * LDSPITCH + k + kk];
        const int kr = k + kk;
        v2f b;
        b.x = We[(size_t)kr * D + colg];
        b.y = We[(size_t)(kr + 1) * D + colg];
        acc0 = __builtin_amdgcn_wmma_f32_16x16x4_f32(false, a0, false, b, (short)0, acc0, false, false);
        acc1 = __builtin_amdgcn_wmma_f32_16x16x4_f32(false, a1, false, b, (short)0, acc1, false, false);
    }

    const float wat = Watt[colg];
    const float vbe = be[colg];
#pragma unroll
    for (int t = 0; t < 2; ++t) {
        const v8f acc = t ? acc1 : acc0;
        float apart[8];
#pragma unroll
        for (int v = 0; v < 8; ++v) {
            const int row = t * 16 + half * 8 + v;
            float val = acc[v] + vbe
                      + Ps1[(size_t)s_idx[row] * D + colg]
                      + Pr [(size_t)r_idx[row] * D + colg];
            edges_new[(size_t)(e0 + row) * D + colg] = val;
            // partial of edges_new[row] . W_att over this 16-col slice
            float p = val * wat;
            p += __shfl_xor(p, 1, 32);
            p += __shfl_xor(p, 2, 32);
            p += __shfl_xor(p, 4, 32);
            p += __shfl_xor(p, 8, 32);     // sum within each 16-lane half
            apart[v] = p;
        }
        if (l15 == 0) {
#pragma unroll
            for (int v = 0; v < 8; ++v)
                atomicAdd(&lds_att[t * 16 + half * 8 + v], apart[v]);  // ds_add_f32
        }
    }
    __syncthreads();

    if (ty == 0) {                          // 32 lanes -> 32 edges
        float av = lds_att[tx] + batt[0];
        float lr = av > 0.0f ? av : 0.01f * av;            // leaky_relu(0.01)
        att_out[e0 + tx] = lr;
        atomicMax(&seg_max[r_idx[tx]], enc_f32(lr));       // segment max
    }
}

// --------------------------------------- exp + denominator (segment sum) ----
__global__ __launch_bounds__(256) void gnn_exp(
    float* __restrict__ att_io, const int* __restrict__ receivers,
    const unsigned* __restrict__ seg_max, float* __restrict__ denom, int E)
{
    int e = blockIdx.x * blockDim.x + threadIdx.x;
    if (e >= E) return;
    const float m = dec_f32(seg_max[receivers[e]]);
    const float v = expf(att_io[e] - m);
    att_io[e] = v;                                   // reuse buffer as expv
    atomicAdd(&denom[receivers[e]], v);
}

// ----------------------------------- weighted message scatter-sum ----
// One wave per edge: 32 lanes x float4 cover 128 cols. L2-resident targets.
__global__ __launch_bounds__(256) void gnn_scatter(
    const float* __restrict__ expv, const float* __restrict__ denom,
    const int* __restrict__ senders, const int* __restrict__ receivers,
    const float* __restrict__ Pm, float* __restrict__ nodes_new, int E)
{
    const int wv   = threadIdx.x >> 5;
    const int lane = threadIdx.x & 31;
    const int e = blockIdx.x * 8 + wv;
    if (e >= E) return;
    const int s = senders[e];
    const int r = receivers[e];
    const float w = expv[e] / denom[r];
    const int c = lane * 4;
    const float4 pm = *(const float4*)&Pm[(size_t)s * D + c];
    float* dst = &nodes_new[(size_t)r * D + c];
    atomicAdd(dst + 0, w * pm.x);
    atomicAdd(dst + 1, w * pm.y);
    atomicAdd(dst + 2, w * pm.z);
    atomicAdd(dst + 3, w * pm.w);
}

// -------------------------------------------------------------- launch ----
extern "C" void kernel_launch(void* const* d_in, const int* in_sizes, int n_in,
                              void* d_out, int out_size, void* d_ws, size_t ws_size,
                              hipStream_t stream)
{
    const float* nodes     = (const float*)d_in[0];
    const float* edges     = (const float*)d_in[1];
    const int*   senders   = (const int*)  d_in[2];
    const int*   receivers = (const int*)  d_in[3];
    const float* W_s1 = (const float*)d_in[4];
    const float* b_s1 = (const float*)d_in[5];
    const float* W_r  = (const float*)d_in[6];
    const float* b_r  = (const float*)d_in[7];
    const float* W_e  = (const float*)d_in[8];
    const float* b_e  = (const float*)d_in[9];
    const float* W_att= (const float*)d_in[10];
    const float* b_att= (const float*)d_in[11];
    const float* W_m  = (const float*)d_in[12];
    const float* b_m  = (const float*)d_in[13];

    const int N = in_sizes[0] / D;   // 50000
    const int E = in_sizes[2];       // 800000

    float* nodes_new = (float*)d_out;                       // [N, D]
    float* edges_new = nodes_new + (size_t)N * D;           // [E, D]

    float*    ws    = (float*)d_ws;
    float*    Ps1   = ws;                                   // [N, D]
    float*    Pr    = Ps1 + (size_t)N * D;                  // [N, D]
    float*    Pm    = Pr  + (size_t)N * D;                  // [N, D]
    float*    att   = Pm  + (size_t)N * D;                  // [E] (att -> expv)
    unsigned* seg   = (unsigned*)(att + E);                 // [N]
    float*    denom = (float*)(seg + N);                    // [N]

    // 1) init outputs / segment state
    {
        const int ND = N * D;
        gnn_init<<<(ND + 255) / 256, 256, 0, stream>>>(nodes_new, seg, denom, N, ND);
    }
    // 2) three node projections in one pass
    gnn_node_proj<<<dim3(N / 16, D / 16), 32, 0, stream>>>(
        nodes, W_s1, b_s1, W_r, b_r, W_m, b_m, Ps1, Pr, Pm, N);
    // 3) fused edge GEMM + combine + attention logits + segment max
    gnn_edge<<<dim3(E / 32), dim3(32, 8), 0, stream>>>(
        edges, W_e, b_e, Ps1, Pr, senders, receivers, W_att, b_att,
        edges_new, att, seg, E);
    // 4) exp + denominator
    gnn_exp<<<(E + 255) / 256, 256, 0, stream>>>(att, receivers, seg, denom, E);
    // 5) weighted scatter-sum into nodes_new
    gnn_scatter<<<(E + 7) / 8, 256, 0, stream>>>(
        att, denom, senders, receivers, Pm, nodes_new, E);
}